// _SDEWiredLTCLayer_8864812499233
// MI455X (gfx1250) — compile-verified
//
#include <hip/hip_runtime.h>
#include <cstdint>

// Problem constants (from the reference).
#define B_ 2048
#define I_ 128
#define H_ 256
#define BT 8            // batch rows per block
#define EPS_ 1e-8f
#define LOG2E_ 1.44269504088896340736f

// ---------------------------------------------------------------------------
// gfx1250 async global->LDS copy (CDNA5 ASYNCcnt path).
// LDS destination is lds_off; copies 16 bytes from gaddr (GV mode, saddr=off).
// ---------------------------------------------------------------------------
__device__ __forceinline__ void async_copy_b128(uint32_t lds_off, const void* gaddr) {
    asm volatile("global_load_async_to_lds_b128 %0, %1, off"
                 :: "v"(lds_off), "v"(gaddr)
                 : "memory");
}

__device__ __forceinline__ void wait_async_zero() {
    asm volatile("s_wait_asynccnt 0x0" ::: "memory");
}

// sigmoid(t) = 1 / (1 + 2^(-t*log2(e)))  -> one v_exp_f32 + one v_rcp_f32
__device__ __forceinline__ float fast_sigmoid(float t) {
    float e = __builtin_amdgcn_exp2f(-t * LOG2E_);
    return __builtin_amdgcn_rcpf(1.0f + e);
}

__global__ __launch_bounds__(256, 2)
void ltc_cell_kernel(const float* __restrict__ state,       // [B,H]
                     const float* __restrict__ input,       // [B,I]
                     const float* __restrict__ gleak,       // [H]
                     const float* __restrict__ vleak,       // [H]
                     const float* __restrict__ cm,          // [H]
                     const float* __restrict__ input_w,     // [I,H]
                     const float* __restrict__ input_sigma, // [I,H]
                     const float* __restrict__ input_mu,    // [I,H]
                     const float* __restrict__ input_erev,  // [I,H]
                     const float* __restrict__ w,           // [H,H]
                     const float* __restrict__ sigma,       // [H,H]
                     const float* __restrict__ mu,          // [H,H]
                     const float* __restrict__ erev,        // [H,H]
                     float* __restrict__ out)               // [B,H]
{
    __shared__ float s_in[BT][I_];   // 8*128*4 = 4 KB
    __shared__ float s_st[BT][H_];   // 8*256*4 = 8 KB

    const int h  = threadIdx.x;          // 0..255, one h per thread
    const int b0 = blockIdx.x * BT;      // first batch row of this block

    // ---- async-stage input tile: BT*I_ floats = 4KB = 256 x b128 ----------
    {
        uint32_t loff = (uint32_t)(uintptr_t)(&s_in[0][0]) + (uint32_t)h * 16u;
        const char* g = (const char*)(input + (size_t)b0 * I_) + (size_t)h * 16u;
        async_copy_b128(loff, g);
    }
    // ---- async-stage state tile: BT*H_ floats = 8KB = 512 x b128 ----------
    {
        uint32_t base = (uint32_t)(uintptr_t)(&s_st[0][0]);
        const char* g = (const char*)(state + (size_t)b0 * H_);
        async_copy_b128(base + (uint32_t)h * 16u,           g + (size_t)h * 16u);
        async_copy_b128(base + (uint32_t)(h + 256) * 16u,   g + (size_t)(h + 256) * 16u);
    }
    wait_async_zero();
    __syncthreads();

    float num[BT], den[BT];
#pragma unroll
    for (int b = 0; b < BT; ++b) { num[b] = 0.0f; den[b] = 0.0f; }

    // ---- input-side synapses: contract over i (params coalesced over h) ---
    for (int i = 0; i < I_; ++i) {
        const int idx = i * H_ + h;
        const float iw  = fmaxf(input_w[idx], 0.0f);
        const float isg = input_sigma[idx];
        const float imu = input_mu[idx];
        const float ier = input_erev[idx];
#pragma unroll
        for (int b = 0; b < BT; ++b) {
            const float t   = isg * (s_in[b][i] - imu);   // LDS broadcast read
            const float act = iw * fast_sigmoid(t);
            num[b] = fmaf(act, ier, num[b]);
            den[b] += act;
        }
    }

    // ---- recurrent synapses: contract over j ------------------------------
    for (int j = 0; j < H_; ++j) {
        const int idx = j * H_ + h;
        const float ww = fmaxf(w[idx], 0.0f);
        const float sg = sigma[idx];
        const float m  = mu[idx];
        const float er = erev[idx];
#pragma unroll
        for (int b = 0; b < BT; ++b) {
            const float t   = sg * (s_st[b][j] - m);      // LDS broadcast read
            const float act = ww * fast_sigmoid(t);
            num[b] = fmaf(act, er, num[b]);
            den[b] += act;
        }
    }

    // ---- leak + normalize --------------------------------------------------
    const float gl     = fmaxf(gleak[h], 0.0f);
    const float vl     = vleak[h];
    const float inv_cm = 1.0f / (fmaxf(cm[h], 0.0f) + EPS_);
#pragma unroll
    for (int b = 0; b < BT; ++b) {
        const float st   = s_st[b][h];
        const float numr = fmaf(gl, vl, num[b]) - (gl + den[b]) * st;
        out[(size_t)(b0 + b) * H_ + h] = numr * inv_cm;
    }
}

extern "C" void kernel_launch(void* const* d_in, const int* in_sizes, int n_in,
                              void* d_out, int out_size, void* d_ws, size_t ws_size,
                              hipStream_t stream) {
    (void)in_sizes; (void)n_in; (void)out_size; (void)d_ws; (void)ws_size;
    const float* state       = (const float*)d_in[0];
    const float* input       = (const float*)d_in[1];
    const float* gleak       = (const float*)d_in[2];
    const float* vleak       = (const float*)d_in[3];
    const float* cm          = (const float*)d_in[4];
    const float* input_w     = (const float*)d_in[5];
    const float* input_sigma = (const float*)d_in[6];
    const float* input_mu    = (const float*)d_in[7];
    const float* input_erev  = (const float*)d_in[8];
    const float* w           = (const float*)d_in[9];
    const float* sigma       = (const float*)d_in[10];
    const float* mu          = (const float*)d_in[11];
    const float* erev        = (const float*)d_in[12];
    float* out = (float*)d_out;

    dim3 grid(B_ / BT);   // 256 blocks
    dim3 block(256);      // 8 wave32 waves, one h per lane
    ltc_cell_kernel<<<grid, block, 0, stream>>>(state, input, gleak, vleak, cm,
                                                input_w, input_sigma, input_mu, input_erev,
                                                w, sigma, mu, erev, out);
}